// GHMR_43473658970226
// MI455X (gfx1250) — compile-verified
//
#include <hip/hip_runtime.h>

// GHM-R loss, MI455X (gfx1250, wave32).
// Pass 1: stream 192 MB once (HBM-bound, ~8.2 us @ 23.3 TB/s) producing
//         per-block per-bin (sum_loss, count) partials via LDS ds_add atomics.
//         Data staged through LDS with triple-buffered async b128 loads
//         (global_load_async_to_lds_b128 / s_wait_asynccnt).
// Pass 2: tiny deterministic finalize:  out = (1/n) * sum_b sum[b]/cnt[b].
//         ('tot' cancels algebraically.)

#define GHMR_BINS 10
#define GHMR_TPB  256
#define GHMR_REPL 16
#define GHMR_MU   0.02f
#define GHMR_MAXG 2048

#if defined(__HIP_DEVICE_COMPILE__) && defined(__gfx1250__) && \
    __has_builtin(__builtin_amdgcn_global_load_async_to_lds_b128) && \
    __has_builtin(__builtin_amdgcn_s_wait_asynccnt)
#define GHMR_ASYNC 1
#else
#define GHMR_ASYNC 0
#endif

#if GHMR_ASYNC
// Builtin signature (probe-verified via clang diagnostic): param0 = AS(1)
// pointer to __vector_size__(16) int, param1 = AS(3) pointer, two imm ints.
typedef int ghmr_v4i __attribute__((__vector_size__(16)));
typedef __attribute__((address_space(1))) ghmr_v4i ghmr_gv4;
typedef __attribute__((address_space(3))) ghmr_v4i ghmr_lv4;

__device__ __forceinline__ void ghmr_async_ld16(const float4* gsrc, float4* lds_dst) {
  // VADDR = per-lane global address, VDST = per-lane LDS byte address.
  __builtin_amdgcn_global_load_async_to_lds_b128(
      (ghmr_gv4*)gsrc, (ghmr_lv4*)lds_dst, /*offset=*/0, /*cpol=*/0);
}
template <int N>
__device__ __forceinline__ void ghmr_wait_async() {
  __builtin_amdgcn_s_wait_asynccnt(N);  // ASYNCcnt <= N (loads complete in order)
}
#endif

__device__ __forceinline__ void ghmr_accum(float p, float t, float w,
                                           float* ssum, unsigned* scnt, int rep) {
  if (w > 0.0f) {                       // valid = label_weight > 0
    float d  = p - t;
    float r  = sqrtf(fmaf(d, d, GHMR_MU * GHMR_MU));  // r >= mu > 0
    float ls = r - GHMR_MU;             // smooth-L1 loss
    float g  = fabsf(d) / r;            // |diff|/sqrt(diff^2+mu^2) in [0,1]
    int   b  = (int)(g * 10.0f);        // floor for g >= 0
    b = (b > GHMR_BINS - 1) ? (GHMR_BINS - 1) : b;  // g==1.0 rounding corner
    atomicAdd(&ssum[rep * GHMR_BINS + b], ls);      // ds_add_f32
    atomicAdd(&scnt[rep * GHMR_BINS + b], 1u);      // ds_add_u32
  }
}

// All loop-trip parameters are host-precomputed 32-bit scalars (element counts
// fit easily: nElem = 16M, M4 = 4M, T <= 524288) -> no 64-bit s-division.
__global__ void __launch_bounds__(GHMR_TPB)
ghmr_partial_kernel(const float* __restrict__ pred,
                    const float* __restrict__ targ,
                    const float* __restrict__ lw,
                    float* __restrict__ ws_sum,
                    unsigned* __restrict__ ws_cnt,
                    int nElem, int M4, int T, int nfull) {
  __shared__ float    ssum[GHMR_REPL * GHMR_BINS];   // 16-way replicated to cut
  __shared__ unsigned scnt[GHMR_REPL * GHMR_BINS];   // same-address DS serialization
  const int tid = threadIdx.x;
  for (int i = tid; i < GHMR_REPL * GHMR_BINS; i += GHMR_TPB) {
    ssum[i] = 0.0f;
    scnt[i] = 0u;
  }
  __syncthreads();

  const int rep  = tid & (GHMR_REPL - 1);
  const int gtid = (int)blockIdx.x * GHMR_TPB + tid;

  const float4* p4 = (const float4*)pred;
  const float4* t4 = (const float4*)targ;
  const float4* w4 = (const float4*)lw;

#if GHMR_ASYNC
  // Triple-buffered async staging: 2 stages (6 b128 async ops/wave) in flight.
  // Each lane writes/reads only its own LDS slot -> no workgroup barrier needed,
  // only s_wait_asynccnt. No divergence in this loop, so EXEC is all-ones and
  // per-wave ASYNCcnt accounting is exact.
  __shared__ float4 sP[3][GHMR_TPB];
  __shared__ float4 sT[3][GHMR_TPB];
  __shared__ float4 sW[3][GHMR_TPB];
  if (nfull > 0) {
    ghmr_async_ld16(p4 + gtid, &sP[0][tid]);
    ghmr_async_ld16(t4 + gtid, &sT[0][tid]);
    ghmr_async_ld16(w4 + gtid, &sW[0][tid]);
  }
  if (nfull > 1) {
    ghmr_async_ld16(p4 + gtid + T, &sP[1][tid]);
    ghmr_async_ld16(t4 + gtid + T, &sT[1][tid]);
    ghmr_async_ld16(w4 + gtid + T, &sW[1][tid]);
  }
  int cons   = 0;                // buffer consumed this iteration (k % 3)
  int i_pref = gtid + 2 * T;     // element index of stage k+2
  for (int k = 0; k < nfull; ++k) {
    if (k + 2 < nfull) {
      const int bi = (cons == 0) ? 2 : cons - 1;  // (k+2)%3 == (k-1)%3
      ghmr_async_ld16(p4 + i_pref, &sP[bi][tid]);
      ghmr_async_ld16(t4 + i_pref, &sT[bi][tid]);
      ghmr_async_ld16(w4 + i_pref, &sW[bi][tid]);
      ghmr_wait_async<6>();   // stages k+1,k+2 outstanding -> stage k complete
    } else if (k + 1 < nfull) {
      ghmr_wait_async<3>();
    } else {
      ghmr_wait_async<0>();
    }
    float4 p = sP[cons][tid];
    float4 t = sT[cons][tid];
    float4 w = sW[cons][tid];
    ghmr_accum(p.x, t.x, w.x, ssum, scnt, rep);
    ghmr_accum(p.y, t.y, w.y, ssum, scnt, rep);
    ghmr_accum(p.z, t.z, w.z, ssum, scnt, rep);
    ghmr_accum(p.w, t.w, w.w, ssum, scnt, rep);
    cons = (cons == 2) ? 0 : cons + 1;
    i_pref += T;
  }
#else
  for (int k = 0; k < nfull; ++k) {
    const int i = gtid + k * T;
    __builtin_prefetch(&p4[i + 4 * T], 0, 0);  // global_prefetch_b8, far ahead
    float4 p = p4[i];
    float4 t = t4[i];
    float4 w = w4[i];
    ghmr_accum(p.x, t.x, w.x, ssum, scnt, rep);
    ghmr_accum(p.y, t.y, w.y, ssum, scnt, rep);
    ghmr_accum(p.z, t.z, w.z, ssum, scnt, rep);
    ghmr_accum(p.w, t.w, w.w, ssum, scnt, rep);
  }
#endif

  // Tail: fewer than T float4s remain; one guarded iteration covers them.
  {
    const int i = gtid + nfull * T;
    if (i < M4) {
      float4 p = p4[i];
      float4 t = t4[i];
      float4 w = w4[i];
      ghmr_accum(p.x, t.x, w.x, ssum, scnt, rep);
      ghmr_accum(p.y, t.y, w.y, ssum, scnt, rep);
      ghmr_accum(p.z, t.z, w.z, ssum, scnt, rep);
      ghmr_accum(p.w, t.w, w.w, ssum, scnt, rep);
    }
  }
  // Leftover scalars if nElem % 4 != 0 (not hit for (N,4) inputs).
  if (gtid == 0) {
    for (int j = M4 << 2; j < nElem; ++j)
      ghmr_accum(pred[j], targ[j], lw[j], ssum, scnt, rep);
  }

  __syncthreads();
  if (tid < GHMR_BINS) {
    float    s = 0.0f;
    unsigned c = 0u;
    for (int r = 0; r < GHMR_REPL; ++r) {
      s += ssum[r * GHMR_BINS + tid];
      c += scnt[r * GHMR_BINS + tid];
    }
    ws_sum[(size_t)blockIdx.x * GHMR_BINS + tid] = s;
    ws_cnt[(size_t)blockIdx.x * GHMR_BINS + tid] = c;
  }
}

// One wave (wave32): deterministic cross-block reduction in double, then
// out = LOSS_WEIGHT * (1/n) * sum_{b: cnt>0} sum[b]/cnt[b]   ('tot' cancels).
__global__ void __launch_bounds__(32)
ghmr_final_kernel(const float* __restrict__ ws_sum,
                  const unsigned* __restrict__ ws_cnt,
                  float* __restrict__ out, int G) {
  __shared__ double   s_s[32 * GHMR_BINS];
  __shared__ unsigned s_c[32 * GHMR_BINS];
  const int t = threadIdx.x;
  double   a[GHMR_BINS];
  unsigned c[GHMR_BINS];
#pragma unroll
  for (int b = 0; b < GHMR_BINS; ++b) { a[b] = 0.0; c[b] = 0u; }
  for (int g = t; g < G; g += 32) {
#pragma unroll
    for (int b = 0; b < GHMR_BINS; ++b) {
      a[b] += (double)ws_sum[(size_t)g * GHMR_BINS + b];
      c[b] += ws_cnt[(size_t)g * GHMR_BINS + b];
    }
  }
#pragma unroll
  for (int b = 0; b < GHMR_BINS; ++b) {
    s_s[t * GHMR_BINS + b] = a[b];
    s_c[t * GHMR_BINS + b] = c[b];
  }
  __syncthreads();
  if (t == 0) {
    double total = 0.0;
    int    n     = 0;
    for (int b = 0; b < GHMR_BINS; ++b) {
      double   sb = 0.0;
      unsigned cb = 0u;
      for (int l = 0; l < 32; ++l) {
        sb += s_s[l * GHMR_BINS + b];
        cb += s_c[l * GHMR_BINS + b];
      }
      if (cb > 0u) {
        n += 1;
        total += sb / (double)cb;
      }
    }
    out[0] = (n > 0) ? (float)(total / (double)n) : 0.0f;  // LOSS_WEIGHT == 1
  }
}

extern "C" void kernel_launch(void* const* d_in, const int* in_sizes, int n_in,
                              void* d_out, int out_size, void* d_ws, size_t ws_size,
                              hipStream_t stream) {
  const float* pred = (const float*)d_in[0];
  const float* targ = (const float*)d_in[1];
  const float* lw   = (const float*)d_in[2];
  const int nElem = in_sizes[0];  // N*4 floats (16M -> fits in int)

  // Per-block partials: 10 f32 sums + 10 u32 counts = 80 bytes.
  const size_t perBlock = (size_t)GHMR_BINS * (sizeof(float) + sizeof(unsigned));
  size_t maxG = ws_size / perBlock;
  int G = GHMR_MAXG;
  if ((size_t)G > maxG) G = (int)maxG;
  if (G < 1) G = 1;

  const int M4    = nElem >> 2;       // # float4 elements
  const int T     = G * GHMR_TPB;     // total threads in grid
  const int nfull = M4 / T;           // full (all-lanes-valid) pipeline stages

  float*    ws_sum = (float*)d_ws;
  unsigned* ws_cnt = (unsigned*)(ws_sum + (size_t)G * GHMR_BINS);

  ghmr_partial_kernel<<<G, GHMR_TPB, 0, stream>>>(pred, targ, lw, ws_sum, ws_cnt,
                                                  nElem, M4, T, nfull);
  ghmr_final_kernel<<<1, 32, 0, stream>>>(ws_sum, ws_cnt, (float*)d_out, G);
}